// TF_preprocessing_model_justtf_46548855554405
// MI455X (gfx1250) — compile-verified
//
#include <hip/hip_runtime.h>
#include <cstddef>

#define OUTD  260
#define TAPS  6
#define IMH   2048
#define IMW   2048
#define NB    10
#define TILE  16
#define KMAX  128   // band bound: 15*ceil(2048/260) + 7 < 128

typedef __attribute__((ext_vector_type(2))) float v2f;
typedef __attribute__((ext_vector_type(8))) float v8f;

typedef __attribute__((address_space(1))) int g1int;   // global (device) dword
typedef __attribute__((address_space(3))) int l3int;   // LDS dword

#if __has_builtin(__builtin_amdgcn_global_load_async_to_lds_b32)
#define HAVE_ASYNC_LDS 1
#else
#define HAVE_ASYNC_LDS 0
#endif

__device__ __forceinline__ void wait_asynccnt0() {
#if __has_builtin(__builtin_amdgcn_s_wait_asynccnt)
    __builtin_amdgcn_s_wait_asynccnt(0);
#else
    asm volatile("s_wait_asynccnt 0x0" ::: "memory");
#endif
}

__device__ __forceinline__ float lanczos_sinc(float x) {
    // jnp.sinc: sin(pi x)/(pi x), 1 at 0
    float ax = fabsf(x);
    if (ax < 1e-7f) return 1.0f;
    float px = 3.14159265358979323846f * x;
    return sinf(px) / px;
}

__device__ __forceinline__ float src_coord(int o, float s) {
    return ((float)o + 0.5f) * s - 0.5f;
}

__global__ __launch_bounds__(32)
void crop_resize_lanczos_wmma(const float* __restrict__ scores,
                              const float* __restrict__ boxes,
                              const float* __restrict__ img,   // (2048,2048,3)
                              float* __restrict__ out)         // (10,260,260,3)
{
    // LDS: banded weight matrices + staged image block + layout-conversion spill
    __shared__ float wh[TILE][KMAX + 1];      // 16 x Kh   vertical weights (padded stride)
    __shared__ float bw[KMAX][TILE];          // Kw x 16   horizontal weights
    __shared__ float blk[3][KMAX][TILE];      // image band chunk, 3 channels
    __shared__ float spill[TILE][TILE + 1];   // stage-1 result relayout (padded stride)

    const int lane = threadIdx.x;             // wave32
    const int m    = lane & 15;
    const int sel  = lane >> 4;
    const int box  = blockIdx.z;
    const int p0   = blockIdx.y * TILE;       // output row tile
    const int q0   = blockIdx.x * TILE;       // output col tile

    // ---- box metadata (uniform; every lane computes the same values) ----
    const float b0 = boxes[box * 4 + 0], b1 = boxes[box * 4 + 1];
    const float b2 = boxes[box * 4 + 2], b3 = boxes[box * 4 + 3];
    const int y0 = (int)(b0 * (float)IMH);
    const int x0 = (int)(b1 * (float)IMW);
    const int y1 = (int)(b2 * (float)IMH);
    const int x1 = (int)(b3 * (float)IMW);
    const int ch = max(y1 - y0, 1);
    const int cw = max(x1 - x0, 1);
    const float sh = (float)ch / (float)OUTD;
    const float sw = (float)cw / (float)OUTD;

    const bool valid = (scores[0] >= 0.8f) && (scores[box] >= 0.8f) &&
                       (b1 >= 0.0f) && (b3 <= 1.0f);
    const float vmask = valid ? 1.0f : 0.0f;

    // ---- band extents (monotone taps -> band = [first tap of p0, last tap of pl]) ----
    const int pl = min(p0 + TILE - 1, OUTD - 1);
    const int ql = min(q0 + TILE - 1, OUTD - 1);
    const int hfirst = min(max((int)floorf(src_coord(p0, sh)) - 2, 0), ch - 1) + y0;
    const int hlast  = min(max((int)floorf(src_coord(pl, sh)) + 3, 0), ch - 1) + y0;
    const int wfirst = min(max((int)floorf(src_coord(q0, sw)) - 2, 0), cw - 1) + x0;
    const int wlast  = min(max((int)floorf(src_coord(ql, sw)) + 3, 0), cw - 1) + x0;
    const int kh4  = min(((hlast - hfirst + 1) + 3) & ~3, KMAX);   // K, mult of 4
    const int nW   = min(((wlast - wfirst + 1) + 15) >> 4, KMAX / TILE);

    // ---- zero weight matrices ----
    {
        float* w0 = &wh[0][0];
        for (int e = lane; e < TILE * (KMAX + 1); e += 32) w0[e] = 0.0f;
        float* w1 = &bw[0][0];
        for (int e = lane; e < KMAX * TILE; e += 32) w1[e] = 0.0f;
    }
    __syncthreads();

    // ---- build banded Lanczos weight matrices (one lane per output row/col) ----
    if (lane < 16) {
        const int o = min(p0 + lane, OUTD - 1);
        const float src = src_coord(o, sh);
        const int base = (int)floorf(src) - 2;
        float w[TAPS], wsum = 0.0f;
        #pragma unroll
        for (int a = 0; a < TAPS; ++a) {
            float d = src - (float)(base + a);            // |d| <= 3 always
            w[a] = lanczos_sinc(d) * lanczos_sinc(d * (1.0f / 3.0f));
            wsum += w[a];
        }
        const float inv = 1.0f / wsum;
        #pragma unroll
        for (int a = 0; a < TAPS; ++a) {
            int idx = min(max(base + a, 0), ch - 1) + y0; // clamped row
            wh[lane][idx - hfirst] += w[a] * inv;         // duplicates accumulate
        }
    } else {
        const int j = lane - 16;
        const int o = min(q0 + j, OUTD - 1);
        const float src = src_coord(o, sw);
        const int base = (int)floorf(src) - 2;
        float w[TAPS], wsum = 0.0f;
        #pragma unroll
        for (int a = 0; a < TAPS; ++a) {
            float d = src - (float)(base + a);
            w[a] = lanczos_sinc(d) * lanczos_sinc(d * (1.0f / 3.0f));
            wsum += w[a];
        }
        const float inv = 1.0f / wsum;
        #pragma unroll
        for (int a = 0; a < TAPS; ++a) {
            int idx = min(max(base + a, 0), cw - 1) + x0;
            bw[idx - wfirst][j] += w[a] * inv;
        }
    }
    __syncthreads();

    // ---- main banded-GEMM pipeline: out_tile = (Wh x I) x Bw, per 16-col chunk ----
    v8f acc0 = {}; v8f acc1 = {}; v8f acc2 = {};

    for (int cc = 0; cc < nW; ++cc) {
        // stage image band chunk (kh4 rows x 16 cols x 3 channels) into LDS
#if HAVE_ASYNC_LDS
        // gfx1250 async DMA: each lane copies one dword per channel per row-pair
        for (int r = sel; r < kh4; r += 2) {
            const int grow = min(hfirst + r, IMH - 1);            // zero-weight pad rows
            const int gcol = min(wfirst + cc * TILE + m, IMW - 1);
            const float* px = img + ((size_t)grow * IMW + gcol) * 3;
            __builtin_amdgcn_global_load_async_to_lds_b32(
                (g1int*)(px + 0), (l3int*)&blk[0][r][m], 0, 0);
            __builtin_amdgcn_global_load_async_to_lds_b32(
                (g1int*)(px + 1), (l3int*)&blk[1][r][m], 0, 0);
            __builtin_amdgcn_global_load_async_to_lds_b32(
                (g1int*)(px + 2), (l3int*)&blk[2][r][m], 0, 0);
        }
        // while the DMA drains, prefetch next chunk's rows into cache
        if (cc + 1 < nW) {
            const int gcol = min(wfirst + (cc + 1) * TILE, IMW - 1);
            for (int r = lane; r < kh4; r += 32) {
                const int grow = min(hfirst + r, IMH - 1);
                __builtin_prefetch(img + ((size_t)grow * IMW + gcol) * 3, 0, 1);
            }
        }
        wait_asynccnt0();
        __syncthreads();
#else
        for (int r = sel; r < kh4; r += 2) {
            const int grow = min(hfirst + r, IMH - 1);
            const int gcol = min(wfirst + cc * TILE + m, IMW - 1);
            const float* px = img + ((size_t)grow * IMW + gcol) * 3;
            blk[0][r][m] = px[0];
            blk[1][r][m] = px[1];
            blk[2][r][m] = px[2];
        }
        if (cc + 1 < nW) {
            const int gcol = min(wfirst + (cc + 1) * TILE, IMW - 1);
            for (int r = lane; r < kh4; r += 32) {
                const int grow = min(hfirst + r, IMH - 1);
                __builtin_prefetch(img + ((size_t)grow * IMW + gcol) * 3, 0, 1);
            }
        }
        __syncthreads();
#endif

        for (int c = 0; c < 3; ++c) {
            // stage 1: R(16x16) = Wh(16 x kh4) * I(kh4 x 16), K-chunks of 4
            v8f R = {};
            for (int kk = 0; kk < kh4; kk += 4) {
                v2f a, b;
                a.x = wh[m][kk + 2 * sel];          // A: row m, K = kk+2*sel
                a.y = wh[m][kk + 2 * sel + 1];      //          K = kk+2*sel+1
                b.x = blk[c][kk + 2 * sel][m];      // B: row K, col m
                b.y = blk[c][kk + 2 * sel + 1][m];
                R = __builtin_amdgcn_wmma_f32_16x16x4_f32(
                        false, a, false, b, (short)0, R, false, false);
            }
            // relayout R (C/D layout -> A layout) via LDS
            #pragma unroll
            for (int r = 0; r < 8; ++r) spill[r + 8 * sel][m] = R[r];
            __syncthreads();

            // stage 2: acc(16x16) += R(16x16) * Bw_chunk(16x16), four K=4 steps
            v8f* accp = (c == 0) ? &acc0 : (c == 1) ? &acc1 : &acc2;
            #pragma unroll
            for (int kk = 0; kk < 16; kk += 4) {
                v2f a, b;
                a.x = spill[m][kk + 2 * sel];
                a.y = spill[m][kk + 2 * sel + 1];
                b.x = bw[cc * TILE + kk + 2 * sel][m];
                b.y = bw[cc * TILE + kk + 2 * sel + 1][m];
                *accp = __builtin_amdgcn_wmma_f32_16x16x4_f32(
                            false, a, false, b, (short)0, *accp, false, false);
            }
            __syncthreads();   // spill reused by next channel
        }
        __syncthreads();       // blk reused by next chunk
    }

    // ---- store (C/D layout: VGPR r -> M = r+8*sel, N = m), edge-masked ----
    #pragma unroll
    for (int r = 0; r < 8; ++r) {
        const int p = p0 + r + 8 * sel;
        const int q = q0 + m;
        if (p < OUTD && q < OUTD) {
            const size_t o = (((size_t)box * OUTD + p) * OUTD + q) * 3;
            out[o + 0] = acc0[r] * vmask;
            out[o + 1] = acc1[r] * vmask;
            out[o + 2] = acc2[r] * vmask;
        }
    }
}

extern "C" void kernel_launch(void* const* d_in, const int* in_sizes, int n_in,
                              void* d_out, int out_size, void* d_ws, size_t ws_size,
                              hipStream_t stream) {
    const float* scores = (const float*)d_in[0];  // (100,)
    const float* boxes  = (const float*)d_in[1];  // (100,4)
    const float* img    = (const float*)d_in[2];  // (1,2048,2048,3)
    float* out = (float*)d_out;                   // (10,260,260,3)

    dim3 grid((OUTD + TILE - 1) / TILE, (OUTD + TILE - 1) / TILE, NB); // 17x17x10
    crop_resize_lanczos_wmma<<<grid, dim3(32, 1, 1), 0, stream>>>(scores, boxes, img, out);
}